// customGRU_18872086298655
// MI455X (gfx1250) — compile-verified
//
#include <hip/hip_runtime.h>

// ---------------------------------------------------------------------------
// CDNA5 (gfx1250) fp32-WMMA implementation of the 2-layer bidirectional
// LayerNorm-GRU (with the reference's pre-LN r gate and tan nonlinearity).
//
//   T=1024, B=32, IN0=256, H=512, 3H=1536, IN1=1024
//
// Plan:
//   1) xproj_gemm : hoisted input projection GEMM per (layer, direction)
//        [T*B, IN] x [IN, 3H]  ->  xp[T*B, 3H]   (bias fused, time-flip by index)
//        fp32 WMMA 16x16x4, each wave owns a 32x48 C tile.
//   2) gru_scan   : one workgroup (32 waves) per direction. h kept in LDS,
//        Wh streamed from L2 (3MB/dir, resident across all 1024 steps).
//        Per step: WMMA recurrent GEMM (M=32,N=1536,K=512) -> LDS preact
//        buffer -> fused gate combine -> wave32 shfl LayerNorm -> sigmoid/tan
//        -> h update + y store.
//
// Workspace layout (d_ws, fp32): xp[2][T*B*1536] (384MB) | y0[T*B*1024] (128MB)
// Output layout  (d_out, fp32):  y1[T,B,1024] | last_hidden[4,B,512]
// ---------------------------------------------------------------------------

typedef float v2f __attribute__((ext_vector_type(2)));
typedef float v8f __attribute__((ext_vector_type(8)));

#define TT    1024
#define BB    32
#define HH    512
#define G3H   1536
#define LNEPS 1e-5f
#define HS    514    // padded LDS row stride for h  (514*4B % 256B = 8 -> no bank camp)
#define US    1540   // padded LDS row stride for preacts

// V_WMMA_F32_16X16X4_F32: D(16x16 f32) = A(16x4) * B(4x16) + C
// Fragment layout (ISA 7.12.2): A lane l in [0,15]: row M=l, K=k0+{0,1};
// lanes [16,31]: row M=l-16, K=k0+{2,3}. B mirrors with N striped over lanes.
static __device__ __forceinline__ v8f wmma4(v2f a, v2f b, v8f c) {
  return __builtin_amdgcn_wmma_f32_16x16x4_f32(false, a, false, b, (short)0, c,
                                               false, false);
}

// ---------------------------------------------------------------------------
// xproj GEMM:  xp[t*B+b, n] = sum_i X[tsrc*B+b, i] * Wx[n, i] + bx[n]
//   grid = (4, T), block = 256 (8 waves). Wave w: N cols [bx*384+w*48, +48),
//   M rows [t*32, t*32+32) (one t per block row; B=32 keeps tiles t-aligned).
// ---------------------------------------------------------------------------
__global__ __launch_bounds__(256) void xproj_gemm(
    const float* __restrict__ X,    // [T*B, IN]
    const float* __restrict__ Wx,   // [3H, IN]   (this direction)
    const float* __restrict__ bx,   // [3H]
    float* __restrict__ xp,         // [T*B, 3H]  (scan-time order)
    int IN, int flip) {
  const int lane = threadIdx.x & 31;
  const int wave = threadIdx.x >> 5;
  const int half = lane >> 4;   // lane group: K offset 0 or 2
  const int l16  = lane & 15;
  const int n0   = blockIdx.x * 384 + wave * 48;
  const int t    = blockIdx.y;
  const int tsrc = flip ? (TT - 1 - t) : t;

  const float* a0p = X + (size_t)(tsrc * BB + l16) * IN + half * 2;
  const float* a1p = a0p + (size_t)16 * IN;
  const float* b0p = Wx + (size_t)(n0 + l16) * IN + half * 2;
  const float* b1p = b0p + (size_t)16 * IN;
  const float* b2p = b0p + (size_t)32 * IN;

  v8f acc[2][3] = {};
  for (int k = 0; k < IN; k += 4) {
    v2f a0 = *(const v2f*)(a0p + k);
    v2f a1 = *(const v2f*)(a1p + k);
    v2f b0 = *(const v2f*)(b0p + k);
    v2f b1 = *(const v2f*)(b1p + k);
    v2f b2 = *(const v2f*)(b2p + k);
    acc[0][0] = wmma4(a0, b0, acc[0][0]);
    acc[1][0] = wmma4(a1, b0, acc[1][0]);
    acc[0][1] = wmma4(a0, b1, acc[0][1]);
    acc[1][1] = wmma4(a1, b1, acc[1][1]);
    acc[0][2] = wmma4(a0, b2, acc[0][2]);
    acc[1][2] = wmma4(a1, b2, acc[1][2]);
  }
  // C/D layout: vgpr v, lanes 0-15 -> M=v, N=lane; lanes 16-31 -> M=v+8.
  for (int i = 0; i < 3; ++i) {
    const int col   = n0 + i * 16 + l16;
    const float bias = bx[col];
    for (int mt = 0; mt < 2; ++mt) {
#pragma unroll
      for (int v = 0; v < 8; ++v) {
        const int mrow = mt * 16 + v + half * 8;
        xp[((size_t)t * BB + mrow) * G3H + col] = acc[mt][i][v] + bias;
      }
    }
  }
}

// ---------------------------------------------------------------------------
// Recurrent scan. grid = 2 (direction), block = 1024 (32 waves).
// Dynamic LDS: h[32][HS] + preact up[32][US]  (~257 KB, <= 320 KB/WGP).
// ---------------------------------------------------------------------------
__global__ __launch_bounds__(1024) void gru_scan(
    const float* __restrict__ xp,    // [2][T*B*3H]
    const float* __restrict__ Wh,    // [2][3H*H]  (row n = gate*H + j_out)
    const float* __restrict__ bh,    // [2][3H]
    const float* __restrict__ lng,   // [2][3H]
    const float* __restrict__ lnb,   // [2][3H]
    float* __restrict__ y,           // [T, B, 2H]
    float* __restrict__ last) {      // [2][B*H]  (this layer: fwd, bwd)
  extern __shared__ float smem[];
  float* hbuf = smem;             // 32 * HS
  float* up   = smem + 32 * HS;   // 32 * US

  const int dir  = blockIdx.x;
  const int tid  = threadIdx.x;
  const int lane = tid & 31;
  const int wave = tid >> 5;
  const int half = lane >> 4;
  const int l16  = lane & 15;

  const float* xp_d = xp + (size_t)dir * TT * BB * G3H;
  const float* Whd  = Wh + (size_t)dir * G3H * HH;
  const float* bhd  = bh + (size_t)dir * G3H;
  const float* gzd  = lng + (size_t)dir * G3H;
  const float* bbd  = lnb + (size_t)dir * G3H;

  for (int i = tid; i < 32 * HS; i += 1024) hbuf[i] = 0.0f;
  __syncthreads();

  // Wave w owns N columns [w*48, w*48+48) of the 1536 recurrent outputs.
  const int nb = wave * 48;
  const float* w0p = Whd + (size_t)(nb + l16) * HH + half * 2;
  const float* w1p = w0p + 16 * HH;
  const float* w2p = w0p + 32 * HH;
  const float* h0p = hbuf + l16 * HS + half * 2;  // A frag rows, M tile 0
  const float* h1p = h0p + 16 * HS;               // A frag rows, M tile 1

  for (int t = 0; t < TT; ++t) {
    // -- up = h @ Wh^T (M=32, N=1536, K=512), fp32 WMMA ----------------------
    v8f acc[2][3] = {};
    for (int k = 0; k < HH; k += 4) {
      v2f a0 = *(const v2f*)(h0p + k);
      v2f a1 = *(const v2f*)(h1p + k);
      v2f b0 = *(const v2f*)(w0p + k);
      v2f b1 = *(const v2f*)(w1p + k);
      v2f b2 = *(const v2f*)(w2p + k);
      acc[0][0] = wmma4(a0, b0, acc[0][0]);
      acc[1][0] = wmma4(a1, b0, acc[1][0]);
      acc[0][1] = wmma4(a0, b1, acc[0][1]);
      acc[1][1] = wmma4(a1, b1, acc[1][1]);
      acc[0][2] = wmma4(a0, b2, acc[0][2]);
      acc[1][2] = wmma4(a1, b2, acc[1][2]);
    }
#pragma unroll
    for (int i = 0; i < 3; ++i)
      for (int mt = 0; mt < 2; ++mt) {
#pragma unroll
        for (int v = 0; v < 8; ++v)
          up[(mt * 16 + v + half * 8) * US + nb + i * 16 + l16] = acc[mt][i][v];
      }
    __syncthreads();

    // -- combine with xproj; keep reference quirks (pre-LN r inside h_hat) ---
    const float* xrow_base = xp_d + (size_t)t * BB * G3H;
    for (int r = 0; r < 16; ++r) {
      const int idx = tid + r * 1024;
      const int b = idx >> 9, jj = idx & 511;
      float* ub = up + b * US;
      const float* xr = xrow_base + (size_t)b * G3H;
      const float uz = ub[jj]        + bhd[jj];
      const float ur = ub[512 + jj]  + bhd[512 + jj];
      const float uh = ub[1024 + jj] + bhd[1024 + jj];
      const float zp = xr[jj] + uz;
      const float rp = xr[512 + jj] + ur;       // PRE-layernorm r
      const float hp = xr[1024 + jj] + rp * uh;
      ub[jj]        = zp;                        // z_pre
      ub[1024 + jj] = hp;                        // hh_pre
    }
    __syncthreads();

    // -- LayerNorm + nonlinearity: 64 rows (b in [0,32) x {z, hh}) -----------
#pragma unroll
    for (int rr = 0; rr < 2; ++rr) {
      const int r   = wave * 2 + rr;
      const int b   = r >> 1;
      const int ish = r & 1;  // 0 = z gate, 1 = hh gate
      float* row = up + b * US + ish * 1024;
      float vals[16];
      float s = 0.0f, sq = 0.0f;
#pragma unroll
      for (int i = 0; i < 16; ++i) {
        const float v = row[lane + 32 * i];
        vals[i] = v;
        s += v;
        sq += v * v;
      }
#pragma unroll
      for (int off = 16; off >= 1; off >>= 1) {
        s  += __shfl_xor(s, off, 32);
        sq += __shfl_xor(sq, off, 32);
      }
      const float mean = s * (1.0f / 512.0f);
      const float var  = sq * (1.0f / 512.0f) - mean * mean;
      const float rstd = rsqrtf(var + LNEPS);
      const float* g  = gzd + (ish ? 2 * HH : 0);
      const float* bb = bbd + (ish ? 2 * HH : 0);
#pragma unroll
      for (int i = 0; i < 16; ++i) {
        const int jj = lane + 32 * i;
        const float v = (vals[i] - mean) * rstd * g[jj] + bb[jj];
        row[jj] = ish ? tanf(v) : (1.0f / (1.0f + expf(-v)));  // tan per ref
      }
    }
    __syncthreads();

    // -- h update + output store --------------------------------------------
    const int out_t = dir ? (TT - 1 - t) : t;
    for (int r = 0; r < 16; ++r) {
      const int idx = tid + r * 1024;
      const int b = idx >> 9, jj = idx & 511;
      const float z  = up[b * US + jj];
      const float hh = up[b * US + 1024 + jj];
      const float ho = hbuf[b * HS + jj];
      const float hn = z * ho + (1.0f - z) * hh;
      hbuf[b * HS + jj] = hn;
      y[((size_t)out_t * BB + b) * (2 * HH) + dir * HH + jj] = hn;
    }
    __syncthreads();
  }

  float* last_d = last + (size_t)dir * BB * HH;
  for (int r = 0; r < 16; ++r) {
    const int idx = tid + r * 1024;
    const int b = idx >> 9, jj = idx & 511;
    last_d[b * HH + jj] = hbuf[b * HS + jj];
  }
}

// ---------------------------------------------------------------------------
extern "C" void kernel_launch(void* const* d_in, const int* in_sizes, int n_in,
                              void* d_out, int out_size, void* d_ws,
                              size_t ws_size, hipStream_t stream) {
  (void)in_sizes; (void)n_in; (void)out_size; (void)ws_size;

  const float* x      = (const float*)d_in[0];
  const float* l0_Wx  = (const float*)d_in[1];
  const float* l0_bx  = (const float*)d_in[2];
  const float* l0_Wh  = (const float*)d_in[3];
  const float* l0_bh  = (const float*)d_in[4];
  const float* l0_lng = (const float*)d_in[5];
  const float* l0_lnb = (const float*)d_in[6];
  const float* l1_Wx  = (const float*)d_in[7];
  const float* l1_bx  = (const float*)d_in[8];
  const float* l1_Wh  = (const float*)d_in[9];
  const float* l1_bh  = (const float*)d_in[10];
  const float* l1_lng = (const float*)d_in[11];
  const float* l1_lnb = (const float*)d_in[12];

  float* y1   = (float*)d_out;                      // [T,B,2H]
  float* last = y1 + (size_t)TT * BB * 2 * HH;      // [4,B,H]

  float* xpw = (float*)d_ws;                        // [2][T*B*3H]
  float* y0  = xpw + 2 * (size_t)TT * BB * G3H;     // [T*B*2H]

  const dim3 ggrid(4, TT);
  const size_t lds = (size_t)(32 * HS + 32 * US) * sizeof(float);

  // ---- layer 0 ----
  xproj_gemm<<<ggrid, 256, 0, stream>>>(x, l0_Wx, l0_bx, xpw, 256, 0);
  xproj_gemm<<<ggrid, 256, 0, stream>>>(x, l0_Wx + (size_t)G3H * 256,
                                        l0_bx + G3H,
                                        xpw + (size_t)TT * BB * G3H, 256, 1);
  gru_scan<<<2, 1024, lds, stream>>>(xpw, l0_Wh, l0_bh, l0_lng, l0_lnb, y0,
                                     last);

  // ---- layer 1 (input = y0, IN=1024) ----
  xproj_gemm<<<ggrid, 256, 0, stream>>>(y0, l1_Wx, l1_bx, xpw, 1024, 0);
  xproj_gemm<<<ggrid, 256, 0, stream>>>(y0, l1_Wx + (size_t)G3H * 1024,
                                        l1_bx + G3H,
                                        xpw + (size_t)TT * BB * G3H, 1024, 1);
  gru_scan<<<2, 1024, lds, stream>>>(xpw, l1_Wh, l1_bh, l1_lng, l1_lnb, y1,
                                     last + 2 * (size_t)BB * HH);
}